// CGBatched_85289460564362
// MI455X (gfx1250) — compile-verified
//
#include <hip/hip_runtime.h>
#include <stdint.h>

// ---------------- problem constants (match reference: B=64, N=1024) --------
constexpr int kN          = 1024;
constexpr int kTileRows   = 16;                    // rows of A per TDM tile
constexpr int kNumTiles   = kN / kTileRows;        // 64 tiles per matvec
constexpr int kTileFloats = kTileRows * kN;        // 16384 floats = 64 KB
constexpr unsigned kTileBytes = kTileFloats * 4u;  // 65536 B
constexpr int kBuffers    = 4;                     // 4-deep LDS tile pipeline

typedef unsigned int u32x4 __attribute__((ext_vector_type(4)));
typedef int          i32x4 __attribute__((ext_vector_type(4)));
typedef int          i32x8 __attribute__((ext_vector_type(8)));
typedef float        v2f   __attribute__((ext_vector_type(2)));
typedef float        v8f   __attribute__((ext_vector_type(8)));

#if defined(__has_builtin)
#if __has_builtin(__builtin_amdgcn_tensor_load_to_lds)
#define USE_TDM 1
#endif
#if __has_builtin(__builtin_amdgcn_wmma_f32_16x16x4_f32)
#define USE_WMMA_RED 1
#endif
#endif

// Low 32 bits of a flat shared-aperture address == wave-relative LDS byte
// offset (ISA 10.2).
__device__ inline unsigned lds_off(const void* p) {
  return (unsigned)(uintptr_t)p;
}

#ifdef USE_TDM
// Issue one TDM load: kTileRows x 1024 cols of f32 from global -> LDS.
__device__ inline void tdm_issue(unsigned lds_addr, const float* gsrc, i32x8 g1) {
  uint64_t ga = (uint64_t)(uintptr_t)gsrc;
  u32x4 g0;
  g0[0] = 1u;                                            // count=1, user mode
  g0[1] = lds_addr;                                      // lds_addr
  g0[2] = (unsigned)ga;                                  // global_addr[31:0]
  g0[3] = (unsigned)((ga >> 32) & 0x1FFFFFFull)          // global_addr[56:32]
        | (2u << 30);                                    // type=2 ("image")
  i32x4 z4 = {0, 0, 0, 0};
#if __clang_major__ >= 23
  i32x8 z8 = {0, 0, 0, 0, 0, 0, 0, 0};
  __builtin_amdgcn_tensor_load_to_lds(g0, g1, z4, z4, z8, 0);
#else
  __builtin_amdgcn_tensor_load_to_lds(g0, g1, z4, z4, 0);
#endif
}
#endif

// wave32 all-lanes sum, result valid in every lane.
// WMMA path: A[m][0]=v_m, A[m][2]=v_{m+16} (A-VGPR0 = lane value, VGPR1 = 0),
// B = ones(4x16)  =>  D[m][n] = v_m + v_{m+16}.  Per-lane sum of the 8 D VGPRs
// gives rows 0..7 (lanes 0-15) / rows 8..15 (lanes 16-31); one shfl_xor(16)
// completes the 32-lane sum.  Every call site has EXEC == all-ones (required
// by WMMA, which is NOT skipped on EXEC==0).
__device__ inline float warp_red(float v) {
#ifdef USE_WMMA_RED
  v2f a;    a[0] = v;     a[1] = 0.f;
  v2f ones; ones[0] = 1.f; ones[1] = 1.f;
  v8f c = {};
  c = __builtin_amdgcn_wmma_f32_16x16x4_f32(false, a, false, ones,
                                            (short)0, c, false, false);
  float s = ((c[0] + c[1]) + (c[2] + c[3])) + ((c[4] + c[5]) + (c[6] + c[7]));
  s += __shfl_xor(s, 16, 32);
  return s;
#else
#pragma unroll
  for (int off = 16; off > 0; off >>= 1) v += __shfl_xor(v, off, 32);
  return v;
#endif
}

// 1024-thread block reduction; every wave redundantly reduces the 32 wave
// partials (keeps EXEC all-ones at the WMMA, no divergent second stage, and
// the result is already broadcast).
__device__ inline float block_red(float v, float* red, int wid, int lane) {
  v = warp_red(v);
  if (lane == 0) red[wid] = v;
  __syncthreads();
  float s = warp_red(red[lane]);
  __syncthreads();           // red[] reusable by the next reduction
  return s;
}

// part[2*i+0] + part[2*i+1] = (A_b * vin)[i].
// TDM streams 16-row tiles of A through a 4-buffer LDS pipeline (<=3 loads in
// flight); wave w computes row (w>>1) of each tile over column half (w&1).
__device__ inline void matvec(const float* __restrict__ Ab,
                              const float* __restrict__ vin,
                              float* __restrict__ part,
                              float* __restrict__ tile,
                              int wid, int lane, int tid, i32x8 g1) {
  __syncthreads();           // vin fully written; previous part reads done

  const int half = wid & 1;  // which 512-column half of the row
  const int rloc = wid >> 1; // local row 0..15 inside a tile

  // Stage this lane's slice of vin into registers (512 cols per half).
  float4 pv[4];
  const float4* vin4 = (const float4*)vin;
#pragma unroll
  for (int c = 0; c < 4; ++c) pv[c] = vin4[half * 128 + lane + 32 * c];

#ifdef USE_TDM
  const unsigned lds0 = lds_off(tile);
  if (wid == 0) {            // uniform scalar branch: TDM ignores EXEC,
    tdm_issue(lds0 + 0u * kTileBytes, Ab + 0 * (size_t)kTileRows * kN, g1);
    tdm_issue(lds0 + 1u * kTileBytes, Ab + 1 * (size_t)kTileRows * kN, g1);
    tdm_issue(lds0 + 2u * kTileBytes, Ab + 2 * (size_t)kTileRows * kN, g1);
  }
#endif

  for (int t = 0; t < kNumTiles; ++t) {
#ifdef USE_TDM
    if (wid == 0) {
      // Tiles complete in issue order; with tiles 0..min(t+2,NT-1) issued,
      // outstanding <= issued-(t+1) guarantees tile t has landed.
      if      (t <  kNumTiles - 2) __builtin_amdgcn_s_wait_tensorcnt(2);
      else if (t == kNumTiles - 2) __builtin_amdgcn_s_wait_tensorcnt(1);
      else                         __builtin_amdgcn_s_wait_tensorcnt(0);
    }
    __syncthreads();         // publish tile t to all waves
#else
    {                        // fallback: cooperative synchronous copy
      const float4* src = (const float4*)(Ab + (size_t)t * kTileFloats);
      float4* dst = (float4*)(tile + (size_t)(t & 3) * kTileFloats);
#pragma unroll
      for (int q = 0; q < 4; ++q) dst[tid + 1024 * q] = src[tid + 1024 * q];
      __syncthreads();
    }
#endif
    const float4* row4 = (const float4*)(tile + (size_t)(t & 3) * kTileFloats +
                                         (size_t)rloc * kN + (size_t)half * 512);
    float s = 0.f;
#pragma unroll
    for (int c = 0; c < 4; ++c) {
      float4 a = row4[lane + 32 * c];
      float4 p = pv[c];
      s += a.x * p.x + a.y * p.y + a.z * p.z + a.w * p.w;
    }
    s = warp_red(s);
    if (lane == 0) part[2 * (t * kTileRows + rloc) + half] = s;
    __syncthreads();         // everyone done reading buffer (t&3)
#ifdef USE_TDM
    // Tile t+3 goes into buffer (t+3)&3, which held tile t-1 (consumed at
    // iteration t-1, barrier-ordered above).
    if (wid == 0 && t + 3 < kNumTiles)
      tdm_issue(lds0 + (unsigned)((t + 3) & 3) * kTileBytes,
                Ab + (size_t)(t + 3) * kTileRows * kN, g1);
#endif
  }
  __syncthreads();           // part[] complete
}

__global__ __launch_bounds__(1024, 1)
void CGBatched_85289460564362_kernel(const float* __restrict__ U,
                                     const float* __restrict__ Bv,
                                     const float* __restrict__ A,
                                     const int* __restrict__ maxiter_p,
                                     float* __restrict__ out) {
  // 4x64KB A tiles + p + row partials + reduction scratch (~275 KB of 320 KB)
  __shared__ __align__(16) float tile_s[kBuffers * kTileFloats];
  __shared__ __align__(16) float p_s[kN];
  __shared__ __align__(16) float part_s[2 * kN];
  __shared__ float red_s[32];

  const int tid  = (int)threadIdx.x;
  const int lane = tid & 31;
  const int wid  = __builtin_amdgcn_readfirstlane(tid >> 5);  // uniform SGPR
  const int b    = (int)blockIdx.x;
  const int maxiter = *maxiter_p;

  const float* Ab = A + (size_t)b * kN * kN;

  // Constant TDM descriptor group 1 (f32, tensor 1024x1024, tile 1024x16)
  i32x8 g1;
  g1[0] = 0x20000;                // data_size=2 (4 bytes); no multicast/pad
  g1[1] = (int)(1024u << 16);     // tensor_dim0[15:0] = 1024
  g1[2] = (int)(1024u << 16);     // tensor_dim0 hi=0 | tensor_dim1[15:0]=1024
  g1[3] = (int)(1024u << 16);     // tensor_dim1 hi=0 | tile_dim0 = 1024
  g1[4] = kTileRows;              // tile_dim1 = 16, tile_dim2 = 0
  g1[5] = 1024;                   // tensor_dim0_stride = 1024 elements
  g1[6] = 0;                      // stride0 hi | tensor_dim1_stride (unused)
  g1[7] = 0;

  float x  = U [(size_t)b * kN + tid];
  float bv = Bv[(size_t)b * kN + tid];

  // r0 = b - A x0 ; p0 = r0
  p_s[tid] = x;
  matvec(Ab, p_s, part_s, tile_s, wid, lane, tid, g1);
  float r    = bv - (part_s[2 * tid] + part_s[2 * tid + 1]);
  float pval = r;
  p_s[tid] = r;                   // safe: matvec's trailing barrier passed
  float rr = block_red(r * r, red_s, wid, lane);

  for (int k = 0; k < maxiter; ++k) {
    matvec(Ab, p_s, part_s, tile_s, wid, lane, tid, g1);
    float ap    = part_s[2 * tid] + part_s[2 * tid + 1];
    float pAp   = block_red(pval * ap, red_s, wid, lane);
    float alpha = rr / pAp;
    x += alpha * pval;
    r -= alpha * ap;
    float rrn  = block_red(r * r, red_s, wid, lane);
    float beta = rrn / rr;
    pval = r + beta * pval;
    p_s[tid] = pval;              // matvec entry barrier orders this vs reads
    rr = rrn;
  }

  out[(size_t)b * kN + tid] = x;
}

extern "C" void kernel_launch(void* const* d_in, const int* in_sizes, int n_in,
                              void* d_out, int out_size, void* d_ws, size_t ws_size,
                              hipStream_t stream) {
  const float* U  = (const float*)d_in[0];   // u  [B,N]
  const float* Bv = (const float*)d_in[1];   // b  [B,N,1]
  const float* A  = (const float*)d_in[2];   // A  [B,N,N]
  const int*   mi = (const int*)d_in[3];     // maxiter (scalar)
  float* out = (float*)d_out;
  (void)n_in; (void)out_size; (void)d_ws; (void)ws_size;

  const int batches = in_sizes[0] / kN;      // 64
  CGBatched_85289460564362_kernel<<<dim3(batches), dim3(kN), 0, stream>>>(
      U, Bv, A, mi, out);
}